// VQC_20134806683979
// MI455X (gfx1250) — compile-verified
//
#include <hip/hip_runtime.h>

typedef __attribute__((ext_vector_type(2))) float v2f;
typedef __attribute__((ext_vector_type(8))) float v8f;

#define WSTR 36   // row stride (dwords) for 32x32 W matrix in LDS (16B aligned, conflict-free)
#define XSTR 36   // row stride (dwords) for per-element 32-float vectors
#define WAVES 8

// out[e][q] = <Z_q> for circuit:  psi = (⊗U3(w)) * CNOTchain * (⊗ v_q(enc))
// v_q = RZ(u)RY(s)RZ(t)|+>.  psi = M x -> realified W[32x32] * X[32]  via WMMA f32 16x16x4.
__global__ __launch_bounds__(256) void vqc_wmma_kernel(
    const float* __restrict__ enc,   // (B,4,3)
    const float* __restrict__ wts,   // (1,4,3)
    float* __restrict__ out,         // (B,4)
    int Btot) {
  __shared__ float Wl[32 * WSTR];
  __shared__ float Xl[WAVES * 32 * XSTR];

  const int tid  = threadIdx.x;
  const int lane = tid & 31;
  const int wv   = tid >> 5;
  const int nloc = lane & 15;
  const int hi   = lane >> 4;   // lane half: 0 or 1

  // ---------- one-time per block: build real-ified circuit matrix W in LDS ----------
  {
    const int i = tid >> 4, j = tid & 15;   // T entry (i,j), 256 threads -> all entries
    float tre = 1.0f, tim = 0.0f;
    #pragma unroll
    for (int q = 0; q < 4; ++q) {
      const float th = wts[q * 3 + 0], ph = wts[q * 3 + 1], lm = wts[q * 3 + 2];
      float c, s, cp, sp, cl, sl;
      __sincosf(0.5f * th, &s, &c);
      __sincosf(ph, &sp, &cp);
      __sincosf(lm, &sl, &cl);
      const int a = (i >> (3 - q)) & 1, b = (j >> (3 - q)) & 1;
      float er, ei;
      if (a == 0) {
        if (b == 0) { er = c;        ei = 0.0f;    }
        else        { er = -cl * s;  ei = -sl * s; }   // -e^{i lm} s
      } else {
        if (b == 0) { er = cp * s;   ei = sp * s;  }   //  e^{i ph} s
        else {
          const float pr = cp * cl - sp * sl, pi = cp * sl + sp * cl;
          er = pr * c;  ei = pi * c;                   //  e^{i(ph+lm)} c
        }
      }
      const float nr = tre * er - tim * ei;
      const float ni = tre * ei + tim * er;
      tre = nr; tim = ni;
    }
    // CNOT-chain permutation: y[j] = x[sigma(j)]; qubit q <-> bit (3-q)
    int b = j;
    b ^= ((b >> 0) & 1) << 3;   // gate (c=3,t=0) applied last in circuit -> innermost
    b ^= ((b >> 1) & 1) << 0;   // (2,3)
    b ^= ((b >> 2) & 1) << 1;   // (1,2)
    b ^= ((b >> 3) & 1) << 2;   // (0,1)
    const int k = b;            // M[i][k] = T[i][j]
    Wl[i * WSTR + k]              =  tre;   // Re M
    Wl[i * WSTR + k + 16]         = -tim;   // -Im M
    Wl[(i + 16) * WSTR + k]       =  tim;   // Im M
    Wl[(i + 16) * WSTR + k + 16]  =  tre;   // Re M
  }
  __syncthreads();

  // ---------- load A fragments (W) into registers: 2 M-tiles x 8 K-steps ----------
  // A 16x4 f32 layout: lane L holds row m=L%16; VGPR0 = A[m][k0], VGPR1 = A[m][k0+1],
  // with k0 = 0 (lanes 0-15) or 2 (lanes 16-31) within each K-step of 4.
  v2f afrag[2][8];
  #pragma unroll
  for (int mt = 0; mt < 2; ++mt)
    #pragma unroll
    for (int kk = 0; kk < 8; ++kk) {
      const int m  = mt * 16 + nloc;
      const int k0 = kk * 4 + hi * 2;
      v2f a;
      a.x = Wl[m * WSTR + k0];
      a.y = Wl[m * WSTR + k0 + 1];
      afrag[mt][kk] = a;
    }

  // ---------- per-lane: compute encoded product state x (16 complex) ----------
  const int base = blockIdx.x * 256 + wv * 32;
  const int e    = base + lane;
  const int ec   = e < Btot ? e : (Btot > 0 ? Btot - 1 : 0);  // branchless clamp keeps EXEC full

  float xr[16], xi[16];
  {
    const float4* ep = reinterpret_cast<const float4*>(enc + (size_t)ec * 12);
    const float4 A0 = ep[0], A1 = ep[1], A2 = ep[2];
    const float ang[12] = {A0.x, A0.y, A0.z, A0.w, A1.x, A1.y, A1.z, A1.w,
                           A2.x, A2.y, A2.z, A2.w};
    float vr[4][2], vi[4][2];
    const float R = 0.70710678118654752f;
    #pragma unroll
    for (int q = 0; q < 4; ++q) {
      const float t = ang[q * 3 + 0], s = ang[q * 3 + 1], u = ang[q * 3 + 2];
      float ct, st, cs, ss, cu, su;
      __sincosf(0.5f * t, &st, &ct);
      __sincosf(0.5f * s, &ss, &cs);
      __sincosf(0.5f * u, &su, &cu);
      const float am = cs - ss, ap = cs + ss;
      const float ar = ct * am, ai = -st * ap;   // RY * RZ(t)|+> component 0
      const float br = ct * ap, bi =  st * am;   // component 1
      vr[q][0] = (cu * ar + su * ai) * R;  vi[q][0] = (cu * ai - su * ar) * R;
      vr[q][1] = (cu * br - su * bi) * R;  vi[q][1] = (cu * bi + su * br) * R;
    }
    float w2r[4], w2i[4];
    #pragma unroll
    for (int a = 0; a < 2; ++a)
      #pragma unroll
      for (int b = 0; b < 2; ++b) {
        w2r[a * 2 + b] = vr[0][a] * vr[1][b] - vi[0][a] * vi[1][b];
        w2i[a * 2 + b] = vr[0][a] * vi[1][b] + vi[0][a] * vr[1][b];
      }
    float w3r[8], w3i[8];
    #pragma unroll
    for (int a = 0; a < 4; ++a)
      #pragma unroll
      for (int b = 0; b < 2; ++b) {
        w3r[a * 2 + b] = w2r[a] * vr[2][b] - w2i[a] * vi[2][b];
        w3i[a * 2 + b] = w2r[a] * vi[2][b] + w2i[a] * vr[2][b];
      }
    #pragma unroll
    for (int a = 0; a < 8; ++a)
      #pragma unroll
      for (int b = 0; b < 2; ++b) {
        xr[a * 2 + b] = w3r[a] * vr[3][b] - w3i[a] * vi[3][b];
        xi[a * 2 + b] = w3r[a] * vi[3][b] + w3i[a] * vr[3][b];
      }
  }

  // stage x into this wave's LDS slice: row = 32 floats [Re x(0..15), Im x(0..15)]
  float* Xw  = &Xl[wv * 32 * XSTR];
  float* row = Xw + lane * XSTR;
  #pragma unroll
  for (int g = 0; g < 4; ++g) {
    float4 a; a.x = xr[g*4+0]; a.y = xr[g*4+1]; a.z = xr[g*4+2]; a.w = xr[g*4+3];
    float4 b; b.x = xi[g*4+0]; b.y = xi[g*4+1]; b.z = xi[g*4+2]; b.w = xi[g*4+3];
    *reinterpret_cast<float4*>(row + g * 4)      = a;
    *reinterpret_cast<float4*>(row + 16 + g * 4) = b;
  }
  // same-wave LDS producer/consumer: LDS ops are in-order per wave; compiler adds dscnt waits.

  // ---------- GEMM psi[32 x 16] = W[32x32] * X[32x16], two N-tiles of 16 elements ----------
  #pragma unroll
  for (int nt = 0; nt < 2; ++nt) {
    const float* bbase = Xw + (nt * 16 + nloc) * XSTR + hi * 2;  // B: column n = lane%16
    v8f cRe = {};  // M-tile 0 -> Re psi
    v8f cIm = {};  // M-tile 1 -> Im psi
    #pragma unroll
    for (int kk = 0; kk < 8; ++kk) {
      v2f b;
      b.x = bbase[kk * 4 + 0];
      b.y = bbase[kk * 4 + 1];
      cRe = __builtin_amdgcn_wmma_f32_16x16x4_f32(false, afrag[0][kk], false, b,
                                                  (short)0, cRe, false, false);
      cIm = __builtin_amdgcn_wmma_f32_16x16x4_f32(false, afrag[1][kk], false, b,
                                                  (short)0, cIm, false, false);
    }
    // probs + signed sums: lane holds psi components i = r + 8*hi for r in 0..7
    float z0 = 0.f, z1 = 0.f, z2 = 0.f, z3 = 0.f;
    #pragma unroll
    for (int r = 0; r < 8; ++r) {
      const int ii = r + hi * 8;
      const float p = cRe[r] * cRe[r] + cIm[r] * cIm[r];
      z0 += ((ii >> 3) & 1) ? -p : p;   // qubit 0 <-> bit 3
      z1 += ((ii >> 2) & 1) ? -p : p;
      z2 += ((ii >> 1) & 1) ? -p : p;
      z3 += ((ii >> 0) & 1) ? -p : p;
    }
    z0 += __shfl_xor(z0, 16, 32);
    z1 += __shfl_xor(z1, 16, 32);
    z2 += __shfl_xor(z2, 16, 32);
    z3 += __shfl_xor(z3, 16, 32);
    const int eo = base + nt * 16 + nloc;
    if (hi == 0 && eo < Btot) {
      float4 o; o.x = z0; o.y = z1; o.z = z2; o.w = z3;
      *reinterpret_cast<float4*>(out + (size_t)eo * 4) = o;
    }
  }
}

extern "C" void kernel_launch(void* const* d_in, const int* in_sizes, int n_in,
                              void* d_out, int out_size, void* d_ws, size_t ws_size,
                              hipStream_t stream) {
  const float* enc = (const float*)d_in[0];   // (B,4,3) float32
  const float* wts = (const float*)d_in[1];   // (1,4,3) float32
  float* out = (float*)d_out;                 // (B,4) float32
  const int Btot = in_sizes[0] / 12;
  if (Btot <= 0) return;
  const int blocks = (Btot + 255) / 256;
  vqc_wmma_kernel<<<blocks, 256, 0, stream>>>(enc, wts, out, Btot);
}